// MLPAttention_55757265436896
// MI455X (gfx1250) — compile-verified
//
#include <hip/hip_runtime.h>

typedef float v2f __attribute__((ext_vector_type(2)));
typedef float v8f __attribute__((ext_vector_type(8)));

#define B_     4
#define LQ_    128
#define LK_    1024
#define DSZ_   512
#define UNITS_ 256
#define MASK_VAL_ (-1.0e6f)

#if __has_builtin(__builtin_amdgcn_tanhf)
#define TANHF(x) __builtin_amdgcn_tanhf(x)
#else
#define TANHF(x) tanhf(x)
#endif

// ---------------------------------------------------------------------------
// GEMM C[M,N] = A[M,K] * W[N,K]^T  via V_WMMA_F32_16X16X4_F32.
// One wave computes one 16x16 tile of C, stepping K by 4.
// TRANS_OUT=false: out[m*N + n]            (qf, row-major)
// TRANS_OUT=true : out[(b*N + n)*seq + m%seq]  (kf transposed: [b][unit][kpos])
// ---------------------------------------------------------------------------
template <bool TRANS_OUT>
__global__ __launch_bounds__(256) void proj_wmma(
    const float* __restrict__ A,   // (M, K) row-major
    const float* __restrict__ W,   // (N, K) row-major (torch Linear weight)
    float* __restrict__ out,
    int M, int K, int N, int seq)
{
    const int lane   = threadIdx.x & 31;
    const int wave   = threadIdx.x >> 5;
    const int tilesN = N >> 4;
    const int tile   = blockIdx.x * 8 + wave;
    const int tile_m = tile / tilesN;
    const int tile_n = tile % tilesN;

    const int n15  = lane & 15;
    const int ko   = (lane < 16) ? 0 : 2;     // K sub-offset per half-wave
    const int arow = tile_m * 16 + n15;       // A-matrix row (M) for this lane
    const int wrow = tile_n * 16 + n15;       // W row == B-matrix column (N)

    const float* aptr = A + (size_t)arow * K + ko;
    const float* wptr = W + (size_t)wrow * K + ko;

    v8f c = {0.f, 0.f, 0.f, 0.f, 0.f, 0.f, 0.f, 0.f};
#pragma unroll 4
    for (int kk = 0; kk < K; kk += 4) {
        v2f av = *(const v2f*)(aptr + kk);    // A[arow][kk+ko .. kk+ko+1]
        v2f bv = *(const v2f*)(wptr + kk);    // B[kk+ko..][wrow] = W[wrow][kk+ko..]
        c = __builtin_amdgcn_wmma_f32_16x16x4_f32(
                false, av, false, bv, (short)0, c, false, false);
    }

    const int mo = (lane < 16) ? 0 : 8;       // C/D: upper lanes hold M+8
#pragma unroll
    for (int r = 0; r < 8; ++r) {
        const int m = tile_m * 16 + mo + r;
        if (TRANS_OUT) {
            const int b    = m / seq;
            const int kpos = m % seq;
            out[((size_t)b * N + (tile_n * 16 + n15)) * seq + kpos] = c[r];
        } else {
            out[(size_t)m * N + tile_n * 16 + n15] = c[r];
        }
    }
}

// ---------------------------------------------------------------------------
// scores[b,q,k] = sum_u v[u] * tanh(qf[b,q,u] + kfT[b,u,k]),  masked.
// One block per (b,q); thread t owns k = 4t..4t+3 (coalesced float4 loads).
// qf[row][u] and v[u] are block-uniform -> scalar loads.
// ---------------------------------------------------------------------------
__global__ __launch_bounds__(256) void scores_tanh_kernel(
    const float* __restrict__ qf,     // (B*LQ, UNITS)
    const float* __restrict__ kfT,    // (B, UNITS, LK)
    const float* __restrict__ vvec,   // (UNITS)
    const int*   __restrict__ valid_len,
    float* __restrict__ scores)       // (B, LQ, LK)
{
    const int bq = blockIdx.x;
    const int b  = bq / LQ_;
    const int t  = threadIdx.x;

    const float* qrow  = qf  + (size_t)bq * UNITS_;
    const float* kbase = kfT + (size_t)b * UNITS_ * LK_ + t * 4;

    float a0 = 0.f, a1 = 0.f, a2 = 0.f, a3 = 0.f;
#pragma unroll 4
    for (int u = 0; u < UNITS_; ++u) {
        const float qv = qrow[u];
        const float vv = vvec[u];
        const float4 kv = *(const float4*)(kbase + (size_t)u * LK_);
        a0 = __builtin_fmaf(vv, TANHF(qv + kv.x), a0);
        a1 = __builtin_fmaf(vv, TANHF(qv + kv.y), a1);
        a2 = __builtin_fmaf(vv, TANHF(qv + kv.z), a2);
        a3 = __builtin_fmaf(vv, TANHF(qv + kv.w), a3);
    }

    const int vl = valid_len[b];
    const int k0 = t * 4;
    float4 r;
    r.x = (k0 + 0 < vl) ? a0 : MASK_VAL_;
    r.y = (k0 + 1 < vl) ? a1 : MASK_VAL_;
    r.z = (k0 + 2 < vl) ? a2 : MASK_VAL_;
    r.w = (k0 + 3 < vl) ? a3 : MASK_VAL_;
    *(float4*)(scores + (size_t)bq * LK_ + k0) = r;
}

// ---------------------------------------------------------------------------
// In-place softmax over last dim (LK=1024). One block (8 waves) per row.
// ---------------------------------------------------------------------------
__global__ __launch_bounds__(256) void softmax_rows(float* __restrict__ scores)
{
    __shared__ float red[8];
    const int t    = threadIdx.x;
    const int lane = t & 31;
    const int wave = t >> 5;
    float* row = scores + (size_t)blockIdx.x * LK_;

    float4 x = *(const float4*)(row + t * 4);

    float m = fmaxf(fmaxf(x.x, x.y), fmaxf(x.z, x.w));
#pragma unroll
    for (int off = 16; off > 0; off >>= 1)
        m = fmaxf(m, __shfl_xor(m, off, 32));
    if (lane == 0) red[wave] = m;
    __syncthreads();
    float gm = red[0];
#pragma unroll
    for (int i = 1; i < 8; ++i) gm = fmaxf(gm, red[i]);
    __syncthreads();

    float4 e;
    e.x = __expf(x.x - gm);
    e.y = __expf(x.y - gm);
    e.z = __expf(x.z - gm);
    e.w = __expf(x.w - gm);

    float s = e.x + e.y + e.z + e.w;
#pragma unroll
    for (int off = 16; off > 0; off >>= 1)
        s += __shfl_xor(s, off, 32);
    if (lane == 0) red[wave] = s;
    __syncthreads();
    float gs = red[0];
#pragma unroll
    for (int i = 1; i < 8; ++i) gs += red[i];

    const float inv = 1.0f / gs;
    e.x *= inv; e.y *= inv; e.z *= inv; e.w *= inv;
    *(float4*)(row + t * 4) = e;
}

// ---------------------------------------------------------------------------
// out[b] = attn[b] (LQ x LK) * value[b] (LK x DSZ)  via WMMA f32 16x16x4.
// value[b] is already K x N row-major == the B-operand orientation.
// ---------------------------------------------------------------------------
__global__ __launch_bounds__(256) void attn_out_wmma(
    const float* __restrict__ attn,   // (B, LQ, LK)
    const float* __restrict__ value,  // (B, LK, DSZ)
    float* __restrict__ out)          // (B, LQ, DSZ)
{
    const int lane      = threadIdx.x & 31;
    const int wave      = threadIdx.x >> 5;
    const int tilesN    = DSZ_ / 16;              // 32
    const int tilesPerB = (LQ_ / 16) * tilesN;    // 256
    const int tile      = blockIdx.x * 8 + wave;
    const int b         = tile / tilesPerB;
    const int tr        = tile % tilesPerB;
    const int tile_m    = tr / tilesN;
    const int tile_n    = tr % tilesN;

    const int n15 = lane & 15;
    const int ko  = (lane < 16) ? 0 : 2;

    const float* aptr = attn  + ((size_t)b * LQ_ + tile_m * 16 + n15) * LK_ + ko;
    const float* vptr = value + (size_t)b * LK_ * DSZ_
                               + (size_t)ko * DSZ_ + tile_n * 16 + n15;

    v8f c = {0.f, 0.f, 0.f, 0.f, 0.f, 0.f, 0.f, 0.f};
#pragma unroll 4
    for (int kk = 0; kk < LK_; kk += 4) {
        v2f av = *(const v2f*)(aptr + kk);
        const float* vp = vptr + (size_t)kk * DSZ_;
        v2f bv;
        bv.x = vp[0];            // B[kk+ko  ][n]
        bv.y = vp[DSZ_];         // B[kk+ko+1][n]
        c = __builtin_amdgcn_wmma_f32_16x16x4_f32(
                false, av, false, bv, (short)0, c, false, false);
    }

    const int mo = (lane < 16) ? 0 : 8;
#pragma unroll
    for (int r = 0; r < 8; ++r) {
        const int m = tile_m * 16 + mo + r;
        out[((size_t)b * LQ_ + m) * DSZ_ + tile_n * 16 + n15] = c[r];
    }
}

// ---------------------------------------------------------------------------
extern "C" void kernel_launch(void* const* d_in, const int* in_sizes, int n_in,
                              void* d_out, int out_size, void* d_ws, size_t ws_size,
                              hipStream_t stream) {
    const float* query     = (const float*)d_in[0];  // (B, LQ, 512)
    const float* key       = (const float*)d_in[1];  // (B, LK, 512)
    const float* value     = (const float*)d_in[2];  // (B, LK, 512)
    const int*   valid_len = (const int*)  d_in[3];  // (B,)
    const float* W_q       = (const float*)d_in[4];  // (256, 512)
    const float* W_k       = (const float*)d_in[5];  // (256, 512)
    const float* vvec      = (const float*)d_in[6];  // (256,)

    float* qf     = (float*)d_ws;                         // B*LQ*UNITS   = 131072
    float* kfT    = qf  + (size_t)B_ * LQ_ * UNITS_;      // B*UNITS*LK   = 1048576
    float* scores = kfT + (size_t)B_ * UNITS_ * LK_;      // B*LQ*LK      = 524288
    float* outp   = (float*)d_out;                        // B*LQ*DSZ

    // qf = query @ W_q^T : M=512, K=512, N=256 -> 512 tiles / 8 waves = 64 blocks
    proj_wmma<false><<<64, 256, 0, stream>>>(query, W_q, qf,
                                             B_ * LQ_, DSZ_, UNITS_, 0);
    // kfT = (key @ W_k^T) transposed per-batch: M=4096 -> 4096 tiles -> 512 blocks
    proj_wmma<true><<<512, 256, 0, stream>>>(key, W_k, kfT,
                                             B_ * LK_, DSZ_, UNITS_, LK_);
    // additive scores + mask: one block per (b,q) row
    scores_tanh_kernel<<<B_ * LQ_, 256, 0, stream>>>(qf, kfT, vvec, valid_len, scores);
    // softmax in-place
    softmax_rows<<<B_ * LQ_, 256, 0, stream>>>(scores);
    // out = attn @ value : 4 * 256 tiles / 8 waves = 128 blocks
    attn_out_wmma<<<128, 256, 0, stream>>>(scores, value, outp);
}